// Winograd2d_1949915153157
// MI455X (gfx1250) — compile-verified
//
#include <hip/hip_runtime.h>

typedef __attribute__((ext_vector_type(2))) float v2f;
typedef __attribute__((ext_vector_type(8))) float v8f;

#if defined(__gfx1250__) && __has_builtin(__builtin_amdgcn_wmma_f32_16x16x4_f32)
#define USE_WMMA_F32X4 1
#else
#define USE_WMMA_F32X4 0
#endif

#define NB 32
#define CIN 64
#define HW 56
#define OC 128
#define NTILE 14
#define TILES_PER_IMG (NTILE * NTILE) /* 196 */
#define MTOT (NB * TILES_PER_IMG)     /* 6272 */
#define MBLK 16
#define NBLK 16
#define KCHUNK 16
#define YPITCH 68 /* padded pitch (floats) for the epilogue exchange buffer */

// t = BT(8x8) * d(8-vector), BT for F(4x4,5x5)
__device__ __forceinline__ void bt8(const float d[8], float t[8]) {
    t[0] = d[0] - 5.25f * d[2] + 5.25f * d[4] - d[6];
    t[1] =  d[1] + d[2] - 4.25f * d[3] - 4.25f * d[4] + d[5] + d[6];
    t[2] = -d[1] + d[2] + 4.25f * d[3] - 4.25f * d[4] - d[5] + d[6];
    t[3] =  0.5f * d[1] + 0.25f * d[2] - 2.5f * d[3] - 1.25f * d[4] + 2.0f * d[5] + d[6];
    t[4] = -0.5f * d[1] + 0.25f * d[2] + 2.5f * d[3] - 1.25f * d[4] - 2.0f * d[5] + d[6];
    t[5] =  2.0f * d[1] + 4.0f * d[2] - 2.5f * d[3] - 5.0f * d[4] + 0.5f * d[5] + d[6];
    t[6] = -2.0f * d[1] + 4.0f * d[2] + 2.5f * d[3] - 5.0f * d[4] - 0.5f * d[5] + d[6];
    t[7] = -d[1] + 5.25f * d[3] - 5.25f * d[5] + d[7];
}

// z = AT(4x8) * y(8-vector)
__device__ __forceinline__ void at4(const float y[8], float z[4]) {
    z[0] = y[0] + y[1] + y[2] + y[3] + y[4] + 8.0f * (y[5] + y[6]);
    z[1] = (y[1] - y[2]) + 2.0f * (y[3] - y[4]) + 4.0f * (y[5] - y[6]);
    z[2] = (y[1] + y[2]) + 4.0f * (y[3] + y[4]) + 2.0f * (y[5] + y[6]);
    z[3] = (y[1] - y[2]) + 8.0f * (y[3] - y[4]) + (y[5] - y[6]) + y[7];
}

__global__ __launch_bounds__(256)
void winograd_fused(const float* __restrict__ x,
                    const float* __restrict__ wgt,
                    const float* __restrict__ bias,
                    float* __restrict__ out) {
    // Dynamic LDS: xt[64p][16c][16t] (64KB) + w[64p][16c][16o] (64KB).
    // After the K loop the whole region is reused as y[16t][16o][68 pitch].
    extern __shared__ float smem[];
    float* lds_xt = smem;
    float* lds_w  = smem + 64 * 256;
    float* lds_y  = smem;

    const int tid  = threadIdx.x;
    const int wave = tid >> 5;
    const int lane = tid & 31;
    const int tb   = blockIdx.x; // 0..391  tile block (16 tiles)
    const int ob   = blockIdx.y; // 0..7    out-channel block (16 oc)

    // --- per-thread tile decode for the input-transform stage ---
    const int t_local  = tid & 15;  // lanes 0..15 -> consecutive tiles (coalescing)
    const int cw_local = tid >> 4;  // channel-within-chunk
    const int tg   = tb * MBLK + t_local; // < 6272 exactly
    const int nimg = tg / TILES_PER_IMG;
    const int rem  = tg % TILES_PER_IMG;
    const int ta   = rem / NTILE;
    const int tbb  = rem % NTILE;
    const int h0   = ta * 4 - 2;
    const int w0   = tbb * 4 - 2;

    v8f acc[8];
#pragma unroll
    for (int q = 0; q < 8; ++q)
#pragma unroll
        for (int r = 0; r < 8; ++r) acc[q][r] = 0.0f;

    for (int chunk = 0; chunk < CIN / KCHUNK; ++chunk) {
        const int c0 = chunk * KCHUNK;

        // ---- Phase A1: input transform, one 8x8 tile per lane ----
        {
            const float* xb = x + (size_t)(nimg * CIN + c0 + cw_local) * HW * HW;
            float U[8][8];
#pragma unroll
            for (int k = 0; k < 8; ++k) {
                const int ww = w0 + k;
                const bool wok = (ww >= 0) & (ww < HW);
                float col[8];
#pragma unroll
                for (int r = 0; r < 8; ++r) {
                    const int hh = h0 + r;
                    const bool ok = wok & (hh >= 0) & (hh < HW);
                    col[r] = ok ? xb[hh * HW + ww] : 0.0f;
                }
                float tt[8];
                bt8(col, tt);
#pragma unroll
                for (int i = 0; i < 8; ++i) U[i][k] = tt[i];
            }
#pragma unroll
            for (int i = 0; i < 8; ++i) {
                float tt[8];
                bt8(U[i], tt);
#pragma unroll
                for (int l = 0; l < 8; ++l)
                    lds_xt[(i * 8 + l) * 256 + cw_local * 16 + t_local] = tt[l];
            }
        }

        // ---- Phase A2: weight chunk -> LDS  w[p][c][o] ----
        {
            const int o_local = tid & 15;
            const int c_local = tid >> 4;
            const float* wr = wgt + (size_t)((ob * NBLK + o_local) * CIN + c0 + c_local) * 64;
            if (chunk + 1 < CIN / KCHUNK)
                __builtin_prefetch(wr + (size_t)KCHUNK * 64, 0, 3); // next chunk's row
#pragma unroll
            for (int p4 = 0; p4 < 16; ++p4) {
                const float4 v = *reinterpret_cast<const float4*>(wr + p4 * 4);
                const int base = c_local * 16 + o_local;
                lds_w[(p4 * 4 + 0) * 256 + base] = v.x;
                lds_w[(p4 * 4 + 1) * 256 + base] = v.y;
                lds_w[(p4 * 4 + 2) * 256 + base] = v.z;
                lds_w[(p4 * 4 + 3) * 256 + base] = v.w;
            }
        }
        __syncthreads();

        // ---- Phase B: GEMM via V_WMMA_F32_16X16X4_F32 ----
        // kk outer / q inner: batch 16 fragment loads, then 8 back-to-back
        // WMMAs on 8 *independent* accumulators (XDL pipelining, one
        // s_wait_dscnt per 8 WMMAs instead of per WMMA).
        {
            const int mrow  = lane & 15;
            const int khalf = (lane >> 4) << 1; // lanes 16..31 hold K=2,3
#pragma unroll
            for (int kk = 0; kk < 4; ++kk) {
                const int cb = kk * 4 + khalf;
                v2f a[8], b[8];
#pragma unroll
                for (int q = 0; q < 8; ++q) {
                    const int p = wave * 8 + q;
                    const float* xa = lds_xt + p * 256;
                    const float* wb = lds_w + p * 256;
                    a[q].x = xa[(cb + 0) * 16 + mrow]; // A: M=tiles, K=chans
                    a[q].y = xa[(cb + 1) * 16 + mrow];
                    b[q].x = wb[(cb + 0) * 16 + mrow]; // B: K=chans, N=oc
                    b[q].y = wb[(cb + 1) * 16 + mrow];
                }
#pragma unroll
                for (int q = 0; q < 8; ++q) {
#if USE_WMMA_F32X4
                    acc[q] = __builtin_amdgcn_wmma_f32_16x16x4_f32(
                        false, a[q], false, b[q], (short)0, acc[q], false, false);
#else
                    // scalar fallback matching the C fragment layout
                    const int p = wave * 8 + q;
                    const float* xa = lds_xt + p * 256;
                    const float* wb = lds_w + p * 256;
#pragma unroll
                    for (int r = 0; r < 8; ++r) {
                        const int m = r + ((lane >> 4) ? 8 : 0);
                        float s = 0.0f;
#pragma unroll
                        for (int c = 0; c < 4; ++c)
                            s += xa[(kk * 4 + c) * 16 + m] * wb[(kk * 4 + c) * 16 + mrow];
                        acc[q][r] += s;
                    }
#endif
                }
            }
        }
        __syncthreads();
    }

    // ---- Phase C1: scatter C fragments -> lds_y[t][o][p] (pitch 68) ----
    // p contiguous per (t,o) so the epilogue can read with ds_load_b128.
    {
        const int mhi = (lane >> 4) ? 8 : 0;
        const int o   = lane & 15;
#pragma unroll
        for (int q = 0; q < 8; ++q) {
            const int p = wave * 8 + q;
#pragma unroll
            for (int r = 0; r < 8; ++r)
                lds_y[((r + mhi) * 16 + o) * YPITCH + p] = acc[q][r];
        }
    }
    __syncthreads();

    // ---- Phase C2: output transform AT * Y * AT^T, bias, store ----
    {
        const int o_local = tid & 15;
        const int t_loc   = tid >> 4;
        const int tg2 = tb * MBLK + t_loc;
        const int n2  = tg2 / TILES_PER_IMG;
        const int r2  = tg2 % TILES_PER_IMG;
        const int a2  = r2 / NTILE;
        const int b2  = r2 % NTILE;

        // 64 values for this (tile, oc): 16 x ds_load_b128 (16B-aligned pitch)
        float Y[8][8];
        const float* yb = lds_y + (t_loc * 16 + o_local) * YPITCH;
#pragma unroll
        for (int v = 0; v < 16; ++v) {
            const float4 f = *reinterpret_cast<const float4*>(yb + v * 4);
            const int j = v >> 1;
            const int k0 = (v & 1) * 4;
            Y[j][k0 + 0] = f.x;
            Y[j][k0 + 1] = f.y;
            Y[j][k0 + 2] = f.z;
            Y[j][k0 + 3] = f.w;
        }

        float U2[4][8];
#pragma unroll
        for (int k = 0; k < 8; ++k) {
            float col[8], z[4];
#pragma unroll
            for (int j = 0; j < 8; ++j) col[j] = Y[j][k];
            at4(col, z);
#pragma unroll
            for (int i = 0; i < 4; ++i) U2[i][k] = z[i];
        }

        const int og = ob * NBLK + o_local;
        const float bv = bias[og];
        float* op = out + ((size_t)(n2 * OC + og) * HW + a2 * 4) * HW + b2 * 4;
#pragma unroll
        for (int i = 0; i < 4; ++i) {
            float z[4];
            at4(U2[i], z);
            const float4 vv = make_float4(z[0] + bv, z[1] + bv, z[2] + bv, z[3] + bv);
            *reinterpret_cast<float4*>(op + i * HW) = vv; // 16B aligned: row pitch 224B
        }
    }
}

extern "C" void kernel_launch(void* const* d_in, const int* in_sizes, int n_in,
                              void* d_out, int out_size, void* d_ws, size_t ws_size,
                              hipStream_t stream) {
    (void)in_sizes; (void)n_in; (void)out_size; (void)d_ws; (void)ws_size;
    const float* x    = (const float*)d_in[0];
    const float* wgt  = (const float*)d_in[1];
    const float* bias = (const float*)d_in[2];
    float* out = (float*)d_out;

    dim3 grid(MTOT / MBLK, OC / NBLK); // (392, 8)
    dim3 block(256);                   // 8 waves (wave32)
    const size_t shmem = 2u * 64u * 256u * sizeof(float); // 128 KB dynamic LDS
    winograd_fused<<<grid, block, shmem, stream>>>(x, wgt, bias, out);
}